// GraphAttention_86346022518984
// MI455X (gfx1250) — compile-verified
//
#include <hip/hip_runtime.h>
#include <hip/hip_bf16.h>

typedef __attribute__((ext_vector_type(2))) float v2f;
typedef __attribute__((ext_vector_type(4))) float v4f;
typedef __attribute__((ext_vector_type(8))) float v8f;

#define NN 8      // batch
#define DD 16     // in/out channel dim (d == o == 16)
#define FF 512    // feature dim
#define KK (DD * FF)        // flattened reduction dim = 8192
#define LN_EPS 1e-5f

// One workgroup per output column g (512 blocks, 256 threads = 8 waves).
// outPre[n,o,g] = scale * sum_k H[n,k] * W[k*8192 + g*16 + o]
//   H[n,k] = h0[n,k]*a0[f(k),g] + h1[n,k]*a1[f(k),g],  f(k) = k & 511.
// Layer 1 passes h1 = h0 and a1col = 0 so the inner loop is branch-free.
// V_WMMA_F32_16X16X4_F32: M = n (rows 0-7 valid; 8-15 hold duplicates that
// are never read, so no masking needed), N = o (16), K = 4 per step.
__global__ __launch_bounds__(256) void contract_wmma_kernel(
    const float* __restrict__ h0, const float* __restrict__ A0m,
    const float* __restrict__ h1, const float* __restrict__ A1m,
    float scale, const float* __restrict__ Wm, float* __restrict__ out,
    int two) {
  __shared__ __align__(16) v2f acol[FF];     // acol[f] = {a0[f,g], a1[f,g]}
  __shared__ float red[8][16][17];           // [wave][M][N], bank-pad

  const int g    = blockIdx.x;
  const int tid  = threadIdx.x;
  const int w    = tid >> 5;                 // wave id 0..7
  const int lane = tid & 31;
  const int half = lane >> 4;                // K-half select (v0: K=0/K=2)
  const int m    = lane & 15;                // row (A) / col (B) index
  const int mr   = m & 7;                    // clamp into valid h rows

  // Stage interleaved a[:,g] columns into LDS.
  for (int i = tid; i < FF; i += 256) {
    v2f t;
    t.x = A0m[i * FF + g];
    t.y = two ? A1m[i * FF + g] : 0.0f;
    acol[i] = t;
  }
  __syncthreads();

  const float* h0p = h0 + mr * KK;
  const float* h1p = h1 + mr * KK;
  const int kbase  = w * 1024 + 2 * half;    // even -> acol v4 load never wraps
  const float* wpc = Wm + (size_t)kbase * (size_t)KK + (size_t)(g * 16 + m);

  // --- software-pipelined K loop: 256 steps of K=4 ---
  // prologue: operands for it = 0
  int kc = kbase;
  v2f h0v = *(const v2f*)(h0p + kc);
  v2f h1v = *(const v2f*)(h1p + kc);
  v4f av  = *(const v4f*)&acol[kc & (FF - 1)];
  float bx = __builtin_nontemporal_load(wpc);
  float by = __builtin_nontemporal_load(wpc + KK);

  v8f c = {};
  for (int it = 0; it < 255; ++it) {
    // compute operands for current step
    v2f Aop;
    Aop.x = h0v.x * av.x + h1v.x * av.y;
    Aop.y = h0v.y * av.z + h1v.y * av.w;
    v2f Bop;
    Bop.x = bx;
    Bop.y = by;

    // issue next step's loads before the WMMA consumes current regs
    kc  += 4;
    wpc += 4 * (size_t)KK;
    h0v = *(const v2f*)(h0p + kc);
    h1v = *(const v2f*)(h1p + kc);
    av  = *(const v4f*)&acol[kc & (FF - 1)];
    bx  = __builtin_nontemporal_load(wpc);
    by  = __builtin_nontemporal_load(wpc + KK);
    __builtin_prefetch(wpc + 4 * KK, 0, 1);  // W one step further ahead

    c = __builtin_amdgcn_wmma_f32_16x16x4_f32(
        false, Aop, false, Bop, (short)0, c, false, false);
  }
  {  // epilogue: last step
    v2f Aop;
    Aop.x = h0v.x * av.x + h1v.x * av.y;
    Aop.y = h0v.y * av.z + h1v.y * av.w;
    v2f Bop;
    Bop.x = bx;
    Bop.y = by;
    c = __builtin_amdgcn_wmma_f32_16x16x4_f32(
        false, Aop, false, Bop, (short)0, c, false, false);
  }

  // C layout: VGPR j holds (M=j, N=lane) lanes 0-15, (M=j+8) lanes 16-31.
#pragma unroll
  for (int j = 0; j < 8; ++j)
    red[w][j + 8 * half][m] = c[j];
  __syncthreads();

  // Cross-wave reduction; only valid batch rows M<8 are written.
  if (tid < 128) {
    const int M = tid >> 4;   // n
    const int N = tid & 15;   // o
    float s = 0.0f;
#pragma unroll
    for (int ww = 0; ww < 8; ++ww) s += red[ww][M][N];
    out[(M * 16 + N) * FF + g] = s * scale;
  }
}

// LayerNorm over the o axis (size 16) of y:(8,16,512), ddof=0, no affine.
__global__ __launch_bounds__(256) void layernorm16_kernel(
    const float* __restrict__ y, float* __restrict__ out) {
  const int t = blockIdx.x * blockDim.x + threadIdx.x;  // one (n,g) per thread
  if (t >= NN * FF) return;
  const int n = t / FF;
  const int g = t - n * FF;
  const float* p = y + n * DD * FF + g;
  float v[16];
  float s = 0.0f;
#pragma unroll
  for (int o = 0; o < 16; ++o) { v[o] = p[o * FF]; s += v[o]; }
  const float mean = s * (1.0f / 16.0f);
  float s2 = 0.0f;
#pragma unroll
  for (int o = 0; o < 16; ++o) { const float d = v[o] - mean; s2 += d * d; }
  const float inv = rsqrtf(s2 * (1.0f / 16.0f) + LN_EPS);
  float* q = out + n * DD * FF + g;
#pragma unroll
  for (int o = 0; o < 16; ++o) q[o * FF] = (v[o] - mean) * inv;
}

extern "C" void kernel_launch(void* const* d_in, const int* in_sizes, int n_in,
                              void* d_out, int out_size, void* d_ws, size_t ws_size,
                              hipStream_t stream) {
  const float* x  = (const float*)d_in[0];            // (8,16,512)
  const float* W  = (const float*)d_in[1];            // (16,512,512,16)
  const float* A  = (const float*)d_in[2];            // (2,512,512)
  const float* A0 = A;
  const float* A1 = A + FF * FF;

  float* y1pre = (float*)d_ws;                        // 65536 floats
  float* h1    = y1pre + NN * DD * FF;                // 65536 floats
  float* y2pre = h1 + NN * DD * FF;                   // 65536 floats
  float* outp  = (float*)d_out;

  // Layer 1: y1 = LN(contract(x, W, A0))   (h1 aliases h0, a1col = 0)
  contract_wmma_kernel<<<FF, 256, 0, stream>>>(x, A0, x, A0, 1.0f, W, y1pre, 0);
  layernorm16_kernel<<<(NN * FF + 255) / 256, 256, 0, stream>>>(y1pre, h1);

  // Layer 2 (fused, single pass over W):
  // y2 = LN(0.5*(contract(x,W,A1) + contract(h1,W,A0)))
  contract_wmma_kernel<<<FF, 256, 0, stream>>>(x, A1, h1, A0, 0.5f, W, y2pre, 1);
  layernorm16_kernel<<<(NN * FF + 255) / 256, 256, 0, stream>>>(y2pre, outp);
}